// EWMA_ridge_model_40200893890672
// MI455X (gfx1250) — compile-verified
//
#include <hip/hip_runtime.h>
#include <math.h>

// ---------------------------------------------------------------------------
// EWMA ridge GMV backtest for MI455X (gfx1250, wave32, WMMA)
//
// Key identity: U diag(1/(ridge+lam)) U^T == (Sigma + ridge*I)^{-1}
// -> no eigendecomposition; per-window SPD Cholesky solve instead.
//
// Gram trick:   Yt^T diag(w) Yt == (sqrt(w).*Yt)^T (sqrt(w).*Yt)
// -> scale rows once at staging time; A and B fragments read the SAME
//    transposed LDS buffer as single aligned ds_load_b64 ops.
// ---------------------------------------------------------------------------

typedef __attribute__((ext_vector_type(2))) float v2f;
typedef __attribute__((ext_vector_type(8))) float v8f;

#define P 128            // assets
#define STEPD 20         // days per step
#define LAGW 24          // lag windows
#define TW (LAGW*STEPD)  // 480 train rows
#define CHUNK 32         // staged rows per LDS chunk (480/32 = 15 chunks)
#define TSTR 34          // transposed LDS stride (even for b64, 34 mod 64 -> all banks)

// workspace layout (floats): [0..479]=w  [480]=wsum  [481]=denom
//                            [512..512+W)=e_all  [512+W..512+2W)=s_all

// ---------------------------------------------------------------- kernel 0 --
__global__ __launch_bounds__(512) void weights_kernel(const float* __restrict__ a,
                                                      float* __restrict__ ws) {
  __shared__ float r1[512];
  __shared__ float r2[512];
  const int t = threadIdx.x;
  const float a0 = a[0];
  float myw = 0.0f;
  if (t < TW) myw = powf(a0, (float)(TW - 1 - t));
  r1[t] = myw;
  r2[t] = myw * myw;
  __syncthreads();
  for (int s = 256; s > 0; s >>= 1) {
    if (t < s) { r1[t] += r1[t + s]; r2[t] += r2[t + s]; }
    __syncthreads();
  }
  const float sumraw = r1[0], sumraw2 = r2[0];
  if (t < TW) ws[t] = myw / sumraw * (float)TW;       // normalized: sum == T
  if (t == 0) {
    ws[480] = (float)TW;                              // wsum after normalization
    ws[481] = 1.0f - sumraw2 / (sumraw * sumraw);     // denom = 1 - sum(w^2)/T^2
  }
}

// ---------------------------------------------------------------- kernel 1 --
__global__ __launch_bounds__(256) void window_kernel(const float* __restrict__ Y,
                                                     const float* __restrict__ logridge,
                                                     const float* __restrict__ ws,
                                                     float* __restrict__ e_all,
                                                     float* __restrict__ s_all) {
  __shared__ float YsT[P * TSTR];           // 17 KB transposed, sqrt(w)-scaled chunk
  __shared__ float tri[(P * (P + 1)) / 2];  // 33 KB packed lower triangle
  __shared__ float sw_s[TW];                // sqrt(w)
  __shared__ float diag_s[P];               // Cholesky diagonal (kept out of tri)
  __shared__ float m_s[P];
  __shared__ float vy[P];                   // forward-solve RHS accumulator
  __shared__ float ys[P];                   // forward-solve result
  __shared__ float vb[P];                   // back-solve RHS accumulator
  __shared__ float vx[P];                   // back-solve result (unnormalized wgt)
  __shared__ float ret_s[STEPD];
  __shared__ float red_s[2];

  const int j    = blockIdx.x;
  const int tid  = threadIdx.x;
  const int lane = tid & 31;
  const int wv   = tid >> 5;            // wave id 0..7 -> M-tile
  const int l16  = lane & 15;
  const int khalf = (lane >> 4) << 1;   // lanes 0-15 -> K 0,1 ; lanes 16-31 -> K 2,3

  // cache sqrt(EWMA weights)
  for (int i = tid; i < TW; i += 256) sw_s[i] = sqrtf(ws[i]);
  __syncthreads();
  const float wsum  = ws[480];
  const float denom = ws[481];
  const float ridge = __expf(logridge[0]);
  const float scale = 1.0f / ((float)TW * denom);
  const float inv_wsum = 1.0f / wsum;

  // accumulators: wave wv owns rows [wv*16, wv*16+16), all 8 N-tiles
  v8f acc[8];
#pragma unroll
  for (int n = 0; n < 8; ++n)
#pragma unroll
    for (int r = 0; r < 8; ++r) acc[n][r] = 0.0f;

  float m_reg = 0.0f;
  const long rowbase = (long)j * STEPD;
  const float* aBase = &YsT[(wv * 16 + l16) * TSTR + khalf];
  const float* bBase = &YsT[l16 * TSTR + khalf];

  for (int t0 = 0; t0 < TW; t0 += CHUNK) {
    __syncthreads();
    // stage CHUNK rows: coalesced float4 global reads, sqrt(w)-scaled,
    // scattered transposed into LDS (YsT[c][r])
    const float4* src = (const float4*)(Y + (size_t)(rowbase + t0) * P);
    for (int i = tid; i < CHUNK * P / 4; i += 256) {
      const int r  = i >> 5;             // row in chunk (P/4 = 32 float4 per row)
      const int c4 = (i & 31) << 2;      // first column
      const float4 vq = src[i];
      const float swt = sw_s[t0 + r];
      YsT[(c4 + 0) * TSTR + r] = vq.x * swt;
      YsT[(c4 + 1) * TSTR + r] = vq.y * swt;
      YsT[(c4 + 2) * TSTR + r] = vq.z * swt;
      YsT[(c4 + 3) * TSTR + r] = vq.w * swt;
    }
    __syncthreads();

    // weighted column sums: m[c] = sum_t w_t*Y[t][c] = sum_t sw_t*Ys[t][c]
    if (tid < P) {
      float am = 0.0f;
      for (int r = 0; r < CHUNK; ++r) am += sw_s[t0 + r] * YsT[tid * TSTR + r];
      m_reg += am;
    }

    // WMMA f32 16x16x4: symmetric Gram, A and B both from scaled YsT
#pragma unroll
    for (int k4 = 0; k4 < CHUNK; k4 += 4) {
      const v2f a = *(const v2f*)(aBase + k4);
#pragma unroll
      for (int nt = 0; nt < 8; ++nt) {
        const v2f b = *(const v2f*)(bBase + nt * 16 * TSTR + k4);
        acc[nt] = __builtin_amdgcn_wmma_f32_16x16x4_f32(
            false, a, false, b, (short)0, acc[nt], false, false);
      }
    }
  }
  __syncthreads();
  if (tid < P) m_s[tid] = m_reg;
  __syncthreads();

  // assemble A = Sigma + ridge*I into packed lower triangle
  // C/D layout: VGPR r -> M = r (lanes 0-15) or r+8 (lanes 16-31); N = lane%16
  const int rofs = (lane >= 16) ? 8 : 0;
#pragma unroll
  for (int nt = 0; nt < 8; ++nt) {
#pragma unroll
    for (int r = 0; r < 8; ++r) {
      const int row = wv * 16 + r + rofs;
      const int col = nt * 16 + l16;
      if (row >= col) {
        float val = (acc[nt][r] - m_s[row] * m_s[col] * inv_wsum) * scale;
        if (row == col) val += ridge;
        tri[(row * (row + 1)) / 2 + col] = val;
      }
    }
  }
  __syncthreads();

  // in-place Cholesky (packed lower); diagonal kept in diag_s to avoid
  // read/write races on the pivot (2 barriers per step)
  for (int k = 0; k < P; ++k) {
    const float dkk = sqrtf(tri[(k * (k + 1)) / 2 + k]);  // read-only here
    const float inv = 1.0f / dkk;
    if (tid == 0) diag_s[k] = dkk;
    const int i = k + 1 + tid;
    float lik = 0.0f;
    if (i < P) {
      const int ci = (i * (i + 1)) / 2 + k;
      lik = tri[ci] * inv;
      tri[ci] = lik;
    }
    __syncthreads();
    if (i < P) {
      const int ro = (i * (i + 1)) / 2;
      for (int jj = k + 1; jj <= i; ++jj)
        tri[ro + jj] -= lik * tri[(jj * (jj + 1)) / 2 + k];
    }
    __syncthreads();
  }

  // forward solve L y = 1 (1 barrier per step; result to ys[], RHS in vy[])
  if (tid < P) vy[tid] = 1.0f;
  __syncthreads();
  for (int k = 0; k < P; ++k) {
    const float yk = vy[k] / diag_s[k];   // all threads compute redundantly
    if (tid == 0) ys[k] = yk;
    const int i = k + 1 + tid;
    if (i < P) vy[i] -= tri[(i * (i + 1)) / 2 + k] * yk;
    __syncthreads();
  }
  // back solve L^T x = y (1 barrier per step)
  if (tid < P) vb[tid] = ys[tid];
  __syncthreads();
  for (int k = P - 1; k >= 0; --k) {
    const float xk = vb[k] / diag_s[k];
    if (tid == 0) vx[k] = xk;
    const int ro = (k * (k + 1)) / 2;
    if (tid < k) vb[tid] -= tri[ro + tid] * xk;  // L^T[i][k] = L[k][i]
    __syncthreads();
  }

  // normalize to GMV weights and evaluate the test window
  if (tid == 0) {
    float sv = 0.0f;
    for (int i = 0; i < P; ++i) sv += vx[i];
    red_s[0] = 1.0f / sv;
  }
  __syncthreads();
  const float inv_sv = red_s[0];
  if (tid < STEPD) {
    const float* yrow = Y + (size_t)((j + LAGW) * STEPD + tid) * P;
    float r = 0.0f;
    for (int c = 0; c < P; ++c) r += yrow[c] * vx[c];
    ret_s[tid] = r * inv_sv;
  }
  __syncthreads();
  if (tid == 0) {
    float e = 0.0f, s2 = 0.0f;
    for (int d = 0; d < STEPD; ++d) { e += ret_s[d]; s2 += ret_s[d] * ret_s[d]; }
    e_all[j] = e;
    s_all[j] = s2 / (float)STEPD;
  }
}

// ---------------------------------------------------------------- kernel 2 --
__global__ __launch_bounds__(256) void finalize_kernel(const float* __restrict__ e_all,
                                                       const float* __restrict__ s_all,
                                                       float* __restrict__ out, int W) {
  __shared__ float re[256];
  __shared__ float rs[256];
  float e = 0.0f, s = 0.0f;
  for (int jj = threadIdx.x; jj < W; jj += 256) { e += e_all[jj]; s += s_all[jj]; }
  re[threadIdx.x] = e;
  rs[threadIdx.x] = s;
  __syncthreads();
  for (int k = 128; k > 0; k >>= 1) {
    if ((int)threadIdx.x < k) { re[threadIdx.x] += re[threadIdx.x + k];
                                rs[threadIdx.x] += rs[threadIdx.x + k]; }
    __syncthreads();
  }
  if (threadIdx.x == 0) {
    const float s_mean = rs[0] / (float)W;
    const float e_mean = re[0] / (float)W;
    const float vol = sqrtf(s_mean * 252.0f);
    const float mu  = e_mean / (float)STEPD * 252.0f;
    out[0] = vol;
    out[1] = mu;
    out[2] = mu / vol;
  }
}

// ------------------------------------------------------------------ launch --
extern "C" void kernel_launch(void* const* d_in, const int* in_sizes, int n_in,
                              void* d_out, int out_size, void* d_ws, size_t ws_size,
                              hipStream_t stream) {
  const float* Y        = (const float*)d_in[0];
  const float* a        = (const float*)d_in[1];
  const float* logridge = (const float*)d_in[2];
  float* out = (float*)d_out;
  float* ws  = (float*)d_ws;

  const int n = in_sizes[0] / P;          // 20480 days
  const int W = n / STEPD - LAGW;         // 1000 windows

  float* e_all = ws + 512;
  float* s_all = ws + 512 + W;

  hipLaunchKernelGGL(weights_kernel, dim3(1), dim3(512), 0, stream, a, ws);
  hipLaunchKernelGGL(window_kernel, dim3(W), dim3(256), 0, stream,
                     Y, logridge, ws, e_all, s_all);
  hipLaunchKernelGGL(finalize_kernel, dim3(1), dim3(256), 0, stream,
                     e_all, s_all, out, W);
}